// DirichletLoss_87368224735836
// MI455X (gfx1250) — compile-verified
//
#include <hip/hip_runtime.h>

// Problem constants (match reference: B=8, N=4096, K=32, r=0.15)
#define BQ   8
#define NQ   4096
#define KNN  32
#define CAP  256               // per-query hit buffer (mean hits ~58, many sigma margin)
#define R2   (0.15f * 0.15f)
#define CH   128               // candidates per async-staged tile
#define NST  (NQ / CH)         // 32 stages
#define SUBS (CH / 16)         // 8 WMMA sub-tiles per stage

typedef float v2f __attribute__((ext_vector_type(2)));
typedef float v8f __attribute__((ext_vector_type(8)));

// Issue one stage's async copies: 1536B of pos (3 x b128 per lane) and 128B of f
// (1 x b32 per lane) from global into LDS. Tracked by ASYNCcnt (4 per wave).
__device__ __forceinline__ void stage_async(const float* __restrict__ gpos,
                                            const float* __restrict__ gf,
                                            float* sp, float* sf, int lane)
{
#pragma unroll
    for (int k = 0; k < 3; ++k) {
        unsigned lds = (unsigned)(uintptr_t)sp + (unsigned)(lane * 16 + k * 512);
        unsigned long long ga =
            (unsigned long long)(uintptr_t)gpos + (unsigned)(lane * 16 + k * 512);
        asm volatile("global_load_async_to_lds_b128 %0, %1, off"
                     :: "v"(lds), "v"(ga) : "memory");
    }
    unsigned ldsf = (unsigned)(uintptr_t)sf + (unsigned)(lane * 4);
    unsigned long long gaf =
        (unsigned long long)(uintptr_t)gf + (unsigned)(lane * 4);
    asm volatile("global_load_async_to_lds_b32 %0, %1, off"
                 :: "v"(ldsf), "v"(gaf) : "memory");
}

// One wave (32 threads) per block; wave owns 16 query points (one WMMA N-tile)
// of one batch. Candidates stream through a double-buffered async LDS stage;
// each 16-candidate sub-tile is one v_wmma_f32_16x16x4_f32 Gram tile.
__global__ __launch_bounds__(32) void dirichlet_tile_kernel(
    const float* __restrict__ pos,   // [B,N,3] f32
    const float* __restrict__ f,     // [B,N]   f32
    float* __restrict__ partial)     // [gridDim.x] per-block variance sums
{
    const int lane = threadIdx.x;     // 0..31 (wave32)
    const int lx   = lane & 15;       // query column / A-candidate id
    const int hi   = lane >> 4;       // 0: D rows 0..7, 1: D rows 8..15
    const int b    = blockIdx.x >> 8; // NQ/16 == 256 query tiles per batch
    const int qbase = (blockIdx.x & 255) * 16;

    __shared__ float2 s_buf[16][CAP];            // (d2,(fq-fc)^2)   32 KB
    __shared__ int    s_cnt[16];
    __shared__ alignas(16) float  s_pos[2][CH * 3];   // staged xyz    3 KB
    __shared__ alignas(16) float  s_f  [2][CH];       // staged f      1 KB
    __shared__ alignas(16) float2 s_nf [2][CH];       // (|c|^2, f)    2 KB

    if (lane < 16) s_cnt[lane] = 0;

    // ---- query setup: every lane loads query (qbase+lx); full xyz in regs
    const size_t pb = (size_t)b * NQ;
    const float* pq = pos + (pb + qbase + lx) * 3;
    const float qx = pq[0], qy = pq[1], qz = pq[2];
    const float fq = f[pb + qbase + lx];
    const float qn2 = qx * qx + qy * qy + qz * qz;

    // B operand (4x16): VGPR0 lanes0-15=K0(x), lanes16-31=K2(z);
    //                   VGPR1 lanes0-15=K1(y), lanes16-31=K3(0)
    v2f Bm;
    Bm.x = hi ? qz : qx;
    Bm.y = hi ? 0.0f : qy;

    // Prologue: async-stage tile 0
    stage_async(pos + pb * 3, f + pb, &s_pos[0][0], &s_f[0][0], lane);
    __syncthreads();

    for (int t = 0; t < NST; ++t) {
        const int buf = t & 1;

        if (t + 1 < NST) {
            stage_async(pos + (pb + (size_t)(t + 1) * CH) * 3,
                        f + pb + (size_t)(t + 1) * CH,
                        &s_pos[(t + 1) & 1][0], &s_f[(t + 1) & 1][0], lane);
            if (t + 2 < NST)   // keep the tile after that warm in L2/WGP$
                __builtin_prefetch(pos + (pb + (size_t)(t + 2) * CH) * 3, 0, 3);
            asm volatile("s_wait_asynccnt 0x4" ::: "memory"); // stage t landed
        } else {
            asm volatile("s_wait_asynccnt 0x0" ::: "memory");
        }

        // Build the (|cand|^2, f_cand) table for this stage: 4 cands per lane
#pragma unroll
        for (int k = 0; k < CH / 32; ++k) {
            const int c = lane + k * 32;
            const float x = s_pos[buf][c * 3 + 0];
            const float y = s_pos[buf][c * 3 + 1];
            const float z = s_pos[buf][c * 3 + 2];
            s_nf[buf][c] = make_float2(x * x + y * y + z * z, s_f[buf][c]);
        }
        __syncthreads();

#pragma unroll
        for (int sub = 0; sub < SUBS; ++sub) {
            const int c0 = sub * 16;
            const float* cp = &s_pos[buf][(c0 + lx) * 3];
            const float cx = cp[0], cy = cp[1], cz = cp[2];

            // A operand (16x4): lanes0-15 K0/K1 (x,y), lanes16-31 K2/K3 (z,0)
            v2f Am;
            Am.x = hi ? cz : cx;
            Am.y = hi ? 0.0f : cy;

            v8f cacc = {0.f, 0.f, 0.f, 0.f, 0.f, 0.f, 0.f, 0.f};
            // D[m][n] = cand_m . query_n
            v8f dot = __builtin_amdgcn_wmma_f32_16x16x4_f32(
                false, Am, false, Bm, (short)0, cacc, false, false);

            // 8 (n2,f) pairs for rows m = c0 + 8*hi + 0..7, as 4 float4 loads
            const float4* nf4 = (const float4*)&s_nf[buf][c0 + 8 * hi];
#pragma unroll
            for (int vv = 0; vv < 4; ++vv) {
                const float4 q = nf4[vv];
                const float d2a = qn2 + q.x - 2.0f * dot[2 * vv];
                if (d2a <= R2) {
                    const float df = fq - q.y;
                    const int slot = atomicAdd(&s_cnt[lx], 1);
                    if (slot < CAP) s_buf[lx][slot] = make_float2(d2a, df * df);
                }
                const float d2b = qn2 + q.z - 2.0f * dot[2 * vv + 1];
                if (d2b <= R2) {
                    const float df = fq - q.w;
                    const int slot = atomicAdd(&s_cnt[lx], 1);
                    if (slot < CAP) s_buf[lx][slot] = make_float2(d2b, df * df);
                }
            }
        }
    }
    __syncthreads();

    // ---- exact top-K selection per query (lane l handles query l), C ~ 58
    float vsum = 0.0f;
    if (lane < 16) {
        int C = s_cnt[lane]; if (C > CAP) C = CAP;
        if (C <= KNN) {
            for (int i = 0; i < C; ++i) vsum += s_buf[lane][i].y;
        } else {
            for (int k = 0; k < KNN; ++k) {       // partial selection sort
                float best = 3.4e38f; int bi = 0;
                for (int i = 0; i < C; ++i) {
                    const float dv = s_buf[lane][i].x;
                    if (dv < best) { best = dv; bi = i; }
                }
                vsum += s_buf[lane][bi].y;
                s_buf[lane][bi].x = 3.4e38f;
            }
        }
    }

    // fixed-tree wave reduction over lanes 0..15 (lanes 16..31 contribute 0)
    for (int off = 8; off; off >>= 1) vsum += __shfl_down(vsum, off, 32);
    if (lane == 0) partial[blockIdx.x] = vsum;
}

// Deterministic final reduction: fixed strided order + fixed LDS tree.
__global__ __launch_bounds__(256) void dirichlet_reduce_kernel(
    const float* __restrict__ partial, float* __restrict__ out, int n)
{
    __shared__ float s[256];
    float a = 0.0f;
    for (int i = threadIdx.x; i < n; i += 256) a += partial[i];
    s[threadIdx.x] = a;
    __syncthreads();
    for (int off = 128; off; off >>= 1) {
        if ((int)threadIdx.x < off) s[threadIdx.x] += s[threadIdx.x + off];
        __syncthreads();
    }
    if (threadIdx.x == 0) out[0] = s[0] * (0.5f / (float)(BQ * NQ));
}

extern "C" void kernel_launch(void* const* d_in, const int* in_sizes, int n_in,
                              void* d_out, int out_size, void* d_ws, size_t ws_size,
                              hipStream_t stream) {
    const float* pos = (const float*)d_in[0];   // [8,4096,3] f32
    const float* f   = (const float*)d_in[1];   // [8,4096]   f32
    float* out       = (float*)d_out;           // scalar f32
    float* partial   = (float*)d_ws;            // 2048 floats scratch

    const int nblocks = BQ * (NQ / 16);         // 2048 single-wave blocks
    dirichlet_tile_kernel<<<nblocks, 32, 0, stream>>>(pos, f, partial);
    dirichlet_reduce_kernel<<<1, 256, 0, stream>>>(partial, out, nblocks);
}